// attention_v2_81879256531394
// MI455X (gfx1250) — compile-verified
//
#include <hip/hip_runtime.h>

typedef __attribute__((ext_vector_type(2))) float v2f;
typedef __attribute__((ext_vector_type(8))) float v8f;

#define A_DIM   32
#define S_LEN   64
#define E_DIM   128
#define K1      2048   // A_DIM * S_LEN
#define OUT_DIM 64
#define K3      256    // 2 * E_DIM

// ---------------------------------------------------------------------------
// Async global->LDS copy (CDNA5): copies 16 bytes from global address g to LDS address l,
// tracked on ASYNCcnt. Generic shared pointers truncate to the LDS offset.
// ---------------------------------------------------------------------------
__device__ __forceinline__ void async_copy_b128(const float* g, float* l)
{
    unsigned lds_off = (unsigned)(unsigned long long)l;
    asm volatile("global_load_async_to_lds_b128 %0, %1, off"
                 :: "v"(lds_off), "v"(g) : "memory");
}

__device__ __forceinline__ void wait_asynccnt0()
{
    asm volatile("s_wait_asynccnt 0x0" ::: "memory");
}

// ---------------------------------------------------------------------------
// Transpose [R,C] -> [C,R] (used to make B fragments contiguous b64 loads).
// ---------------------------------------------------------------------------
__global__ __launch_bounds__(256)
void transpose_kernel(const float* __restrict__ in, float* __restrict__ out,
                      int R, int C)
{
    int idx = blockIdx.x * blockDim.x + threadIdx.x;
    if (idx < R * C) {
        int r = idx / C, c = idx % C;
        out[(size_t)c * R + r] = in[idx];
    }
}

// ---------------------------------------------------------------------------
// Kernel 1: embedded_state[B,128] = state_flat[B,2048] @ W_state[2048,128] + b
// Wt is W_state transposed to [128,2048] so each B fragment is one b64 load:
// lane(l15,h) needs {W[k+2h][n], W[k+1+2h][n]} = Wt[n][k+2h .. k+2h+1].
// ---------------------------------------------------------------------------
__global__ __launch_bounds__(256)
void embed_gemm_kernel(const float* __restrict__ state,
                       const float* __restrict__ Wt,    // [E_DIM, K1]
                       const float* __restrict__ bias,
                       float* __restrict__ C)
{
    const int wave = threadIdx.x >> 5;
    const int lane = threadIdx.x & 31;
    const int half = lane >> 4;
    const int l15  = lane & 15;
    const int m0   = blockIdx.x * 128 + wave * 16;

    v8f zero = {};
    v8f acc[8];
#pragma unroll
    for (int t = 0; t < 8; ++t) acc[t] = zero;

    // A 16x4 f32: lanes 0-15 hold K={k,k+1}, lanes 16-31 hold K={k+2,k+3}
    const float* arow = state + (size_t)(m0 + l15) * K1 + half * 2;
    // B 4x16 f32 via transposed W: one contiguous float2 per fragment
    const float* wrow = Wt + (size_t)l15 * K1 + half * 2;

#pragma unroll 2
    for (int k = 0; k < K1; k += 4) {
        v2f a = *reinterpret_cast<const v2f*>(arow + k);
#pragma unroll
        for (int t = 0; t < 8; ++t) {
            v2f b = *reinterpret_cast<const v2f*>(wrow + (size_t)t * 16 * K1 + k);
            acc[t] = __builtin_amdgcn_wmma_f32_16x16x4_f32(
                false, a, false, b, (short)0, acc[t], false, false);
        }
    }

    // C/D layout: VGPR r -> M = r + 8*half, N = l15
#pragma unroll
    for (int t = 0; t < 8; ++t) {
        const int n = t * 16 + l15;
        const float bn = bias[n];
#pragma unroll
        for (int r = 0; r < 8; ++r) {
            C[(size_t)(m0 + r + 8 * half) * E_DIM + n] = acc[t][r] + bn;
        }
    }
}

// ---------------------------------------------------------------------------
// Kernel 2: attention. One wave per batch row, lane == action index (A=32).
// Embedding gather is done with async global->LDS b128 copies (ASYNCcnt),
// then score + softmax (wave32 shuffles) + weighted sum from LDS.
// ---------------------------------------------------------------------------
__global__ __launch_bounds__(128)
void attention_kernel(const int* __restrict__ action,
                      const float* __restrict__ table,   // [1000,128]
                      const float* __restrict__ W_att,   // [128]
                      const float* __restrict__ b_att,   // [1]
                      float* __restrict__ attended)      // [B,128]
{
    __shared__ float emb[4][A_DIM][E_DIM];   // 64 KB
    __shared__ float wsh[4][A_DIM];

    const int wave = threadIdx.x >> 5;
    const int lane = threadIdx.x & 31;
    const int b    = blockIdx.x * 4 + wave;

    const int act = action[(size_t)b * A_DIM + lane];
    const float* grow = table + (size_t)act * E_DIM;
    float* lrow = &emb[wave][lane][0];

    // async gather: 32 x b128 per lane, per-lane global addr -> per-lane LDS addr
#pragma unroll
    for (int i = 0; i < E_DIM; i += 16) {
#pragma unroll
        for (int j = 0; j < 16; j += 4)
            ; // (unrolled below as single copies of 4 floats)
        async_copy_b128(grow + i,      lrow + i);
        async_copy_b128(grow + i + 4,  lrow + i + 4);
        async_copy_b128(grow + i + 8,  lrow + i + 8);
        async_copy_b128(grow + i + 12, lrow + i + 12);
    }
    wait_asynccnt0();
    __syncthreads();

    // score from LDS row
    float score = b_att[0];
    const float4* wv = reinterpret_cast<const float4*>(W_att);
    const float4* lv = reinterpret_cast<const float4*>(lrow);
#pragma unroll
    for (int i = 0; i < E_DIM / 4; ++i) {
        float4 v  = lv[i];
        float4 w4 = wv[i];
        score += v.x * w4.x + v.y * w4.y + v.z * w4.z + v.w * w4.w;
    }

    // softmax over the 32 actions (one per lane)
    float mx = score;
#pragma unroll
    for (int off = 16; off > 0; off >>= 1)
        mx = fmaxf(mx, __shfl_xor(mx, off, 32));
    float ex  = __expf(score - mx);
    float sum = ex;
#pragma unroll
    for (int off = 16; off > 0; off >>= 1)
        sum += __shfl_xor(sum, off, 32);
    wsh[wave][lane] = ex / sum;

    __syncthreads();

    // each lane produces 4 output channels e0..e0+3
    const int e0 = lane * 4;
    float ax = 0.f, ay = 0.f, az = 0.f, aw = 0.f;
#pragma unroll
    for (int a = 0; a < A_DIM; ++a) {
        const float wa = wsh[wave][a];
        float4 v = *reinterpret_cast<const float4*>(&emb[wave][a][e0]);
        ax += wa * v.x; ay += wa * v.y; az += wa * v.z; aw += wa * v.w;
    }
    float4 o; o.x = ax; o.y = ay; o.z = az; o.w = aw;
    *reinterpret_cast<float4*>(attended + (size_t)b * E_DIM + e0) = o;
}

// ---------------------------------------------------------------------------
// Kernel 3: out[B,64] = concat(es,att)[B,256] @ W_out[256,64] + b_out
// Wt is W_out transposed to [64,256]; B fragments are single b64 loads.
// ---------------------------------------------------------------------------
__global__ __launch_bounds__(256)
void out_gemm_kernel(const float* __restrict__ es,
                     const float* __restrict__ att,
                     const float* __restrict__ Wt,    // [OUT_DIM, K3]
                     const float* __restrict__ bias,  // [64]
                     float* __restrict__ out)         // [B,64]
{
    const int wave = threadIdx.x >> 5;
    const int lane = threadIdx.x & 31;
    const int half = lane >> 4;
    const int l15  = lane & 15;
    const int m0   = blockIdx.x * 128 + wave * 16;

    v8f zero = {};
    v8f acc[4];
#pragma unroll
    for (int t = 0; t < 4; ++t) acc[t] = zero;

    const float* arow0 = es  + (size_t)(m0 + l15) * E_DIM + half * 2;
    const float* arow1 = att + (size_t)(m0 + l15) * E_DIM + half * 2;
    const float* wrow  = Wt + (size_t)l15 * K3 + half * 2;

#pragma unroll 2
    for (int k = 0; k < K3; k += 4) {
        const float* ap = (k < E_DIM) ? (arow0 + k) : (arow1 + (k - E_DIM));
        v2f a = *reinterpret_cast<const v2f*>(ap);
#pragma unroll
        for (int t = 0; t < 4; ++t) {
            v2f b = *reinterpret_cast<const v2f*>(wrow + (size_t)t * 16 * K3 + k);
            acc[t] = __builtin_amdgcn_wmma_f32_16x16x4_f32(
                false, a, false, b, (short)0, acc[t], false, false);
        }
    }

#pragma unroll
    for (int t = 0; t < 4; ++t) {
        const int n = t * 16 + l15;
        const float bn = bias[n];
#pragma unroll
        for (int r = 0; r < 8; ++r) {
            out[(size_t)(m0 + r + 8 * half) * OUT_DIM + n] = acc[t][r] + bn;
        }
    }
}

// ---------------------------------------------------------------------------
extern "C" void kernel_launch(void* const* d_in, const int* in_sizes, int n_in,
                              void* d_out, int out_size, void* d_ws, size_t ws_size,
                              hipStream_t stream)
{
    const float* state   = (const float*)d_in[0];
    const int*   action  = (const int*)  d_in[1];
    const float* W_state = (const float*)d_in[2];
    const float* b_state = (const float*)d_in[3];
    const float* table   = (const float*)d_in[4];
    const float* W_att   = (const float*)d_in[5];
    const float* b_att   = (const float*)d_in[6];
    const float* W_out   = (const float*)d_in[7];
    const float* b_out   = (const float*)d_in[8];
    float* out = (float*)d_out;

    const int B = in_sizes[1] / A_DIM;   // action is [B, 32]

    float* es   = (float*)d_ws;              // [B,128] embedded_state
    float* att  = es  + (size_t)B * E_DIM;   // [B,128] attended
    float* Wt1  = att + (size_t)B * E_DIM;   // [128,2048] W_state^T
    float* Wt3  = Wt1 + (size_t)E_DIM * K1;  // [64,256]   W_out^T

    transpose_kernel<<<(K1 * E_DIM + 255) / 256, 256, 0, stream>>>(W_state, Wt1, K1, E_DIM);
    transpose_kernel<<<(K3 * OUT_DIM + 255) / 256, 256, 0, stream>>>(W_out, Wt3, K3, OUT_DIM);

    embed_gemm_kernel<<<B / 128, 256, 0, stream>>>(state, Wt1, b_state, es);
    attention_kernel <<<B / 4,   128, 0, stream>>>(action, table, W_att, b_att, att);
    out_gemm_kernel  <<<B / 128, 256, 0, stream>>>(es, att, Wt3, b_out, out);
}